// YOLOv9Criterion_21414706937859
// MI455X (gfx1250) — compile-verified
//
#include <hip/hip_runtime.h>
#include <hip/hip_bf16.h>
#include <math.h>

#define B_ 16
#define T_ 60
#define A_ 8400
#define C_ 80
#define EPSF 1e-9f
#define PI_F 3.14159265358979323846f

typedef __attribute__((ext_vector_type(2))) float v2f;
typedef __attribute__((ext_vector_type(8))) float v8f;

// ---------------------------------------------------------------- CIoU (matches reference)
__device__ __forceinline__ float ciou_f(float b1x0, float b1y0, float b1x1, float b1y1,
                                        float b2x0, float b2y0, float b2x1, float b2y1) {
    float xi = fmaxf(b1x0, b2x0), yi = fmaxf(b1y0, b2y0);
    float xa = fminf(b1x1, b2x1), ya = fminf(b1y1, b2y1);
    float inter = fmaxf(xa - xi, 0.f) * fmaxf(ya - yi, 0.f);
    float a1 = (b1x1 - b1x0) * (b1y1 - b1y0);
    float a2 = (b2x1 - b2x0) * (b2y1 - b2y0);
    float iou = inter / (a1 + a2 - inter + EPSF);
    float cx1 = (b1x1 + b1x0) * 0.5f, cy1 = (b1y1 + b1y0) * 0.5f;
    float cx2 = (b2x1 + b2x0) * 0.5f, cy2 = (b2y1 + b2y0) * 0.5f;
    float dx = cx1 - cx2, dy = cy1 - cy2;
    float cent = dx * dx + dy * dy;
    float cw = fmaxf(b1x1, b2x1) - fminf(b1x0, b2x0);
    float ch = fmaxf(b1y1, b2y1) - fminf(b1y0, b2y0);
    float diou = iou - cent / (cw * cw + ch * ch + EPSF);
    float at = atanf((b1x1 - b1x0) / (b1y1 - b1y0 + EPSF))
             - atanf((b2x1 - b2x0) / (b2y1 - b2y0 + EPSF));
    float v = (4.0f / (PI_F * PI_F)) * at * at;
    float alpha = v / (v - iou + 1.0f + EPSF);
    return diou - alpha * v;
}

// fast sigmoid via v_exp_f32 — used IDENTICALLY in k_rowstats and k_match so the
// recomputed tm values stay bit-consistent with the stored thresholds.
__device__ __forceinline__ float sigmoid_fast(float x) {
    return 1.f / (1.f + __expf(-x));
}

__device__ __forceinline__ float tm_value(float iou_c, float sg) {
    float p2 = iou_c * iou_c;
    return p2 * p2 * p2 * sqrtf(sg);   // iou^6 * cls^0.5
}

// bce term at t=0: max(x,0)+log1p(exp(-|x|)); native v_exp_f32/v_log_f32 form
// (log(1+y) vs log1p(y) differs only for |x|>17, abs err < 6e-8 per term)
__device__ __forceinline__ float softplus_bce(float xv) {
    return fmaxf(xv, 0.f) + __logf(1.f + __expf(-fabsf(xv)));
}

__device__ __forceinline__ float blockReduceSum(float v, float* sbuf) {
    int tid = threadIdx.x;
    sbuf[tid] = v;
    __syncthreads();
    for (int s = 128; s >= 1; s >>= 1) {
        if (tid < s) sbuf[tid] += sbuf[tid + s];
        __syncthreads();
    }
    float r = sbuf[0];
    __syncthreads();
    return r;
}

// ---------------------------------------------------------------- Kernel 1: per-(b,t) row stats
// th = 10th-largest of target_matrix row, maxt = row max, maxi = max clipped CIoU.
__global__ __launch_bounds__(256) void k_rowstats(
    const float* __restrict__ target, const float* __restrict__ pcls,
    const float* __restrict__ pbox,   const float* __restrict__ anchors,
    float* __restrict__ th_buf, float* __restrict__ maxt_buf, float* __restrict__ maxi_buf) {
    __shared__ float s_top[256 * 10];
    __shared__ float s_mi[256];
    const int bt = blockIdx.x;             // 0..B*T-1
    const int b  = bt / T_;
    const int tid = threadIdx.x;

    const float* trow = target + (size_t)bt * 5;
    int   c   = (int)trow[0]; if (c < 0) c = 0;
    const float tx0 = trow[1], ty0 = trow[2], tx1 = trow[3], ty1 = trow[4];

    float* mytop = &s_top[tid * 10];
    #pragma unroll
    for (int k = 0; k < 10; ++k) mytop[k] = 0.f;
    float mi = 0.f;

    for (int a = tid; a < A_; a += 256) {
        float ax = anchors[2 * a], ay = anchors[2 * a + 1];
        const float4 pb = *(const float4*)(pbox + ((size_t)b * A_ + a) * 4);
        float iou_c = ciou_f(tx0, ty0, tx1, ty1, pb.x, pb.y, pb.z, pb.w);
        iou_c = fminf(fmaxf(iou_c, 0.f), 1.f);
        mi = fmaxf(mi, iou_c);
        bool gm = (tx0 < ax) & (ax < tx1) & (ty0 < ay) & (ay < ty1);
        if (gm) {
            float x  = pcls[((size_t)b * A_ + a) * C_ + c];
            float tm = tm_value(iou_c, sigmoid_fast(x));
            if (tm > mytop[9]) {                  // insertion into sorted-desc top-10
                int j = 9;
                while (j > 0 && mytop[j - 1] < tm) { mytop[j] = mytop[j - 1]; --j; }
                mytop[j] = tm;
            }
        }
    }
    s_mi[tid] = mi;
    __syncthreads();

    // tree-merge sorted top-10 lists
    for (int str = 128; str >= 1; str >>= 1) {
        if (tid < str) {
            float out[10];
            const float* A0 = &s_top[tid * 10];
            const float* A1 = &s_top[(tid + str) * 10];
            int i = 0, j = 0;
            #pragma unroll
            for (int k = 0; k < 10; ++k) {
                float x = A0[i], y = A1[j];
                if (x >= y) { out[k] = x; ++i; } else { out[k] = y; ++j; }
            }
            #pragma unroll
            for (int k = 0; k < 10; ++k) s_top[tid * 10 + k] = out[k];
            s_mi[tid] = fmaxf(s_mi[tid], s_mi[tid + str]);
        }
        __syncthreads();
    }
    if (tid == 0) {
        th_buf[bt]   = s_top[9];
        maxt_buf[bt] = s_top[0];
        maxi_buf[bt] = s_mi[0];
    }
}

// ---------------------------------------------------------------- Kernel 2: per-(b,a) matching + partials
//   p2 = sum x[b,a,cls]*s ; p3 = sum (1-ciou(pred,align))*s ; ps = sum s
__global__ __launch_bounds__(256) void k_match(
    const float* __restrict__ target, const float* __restrict__ pcls,
    const float* __restrict__ pbox,   const float* __restrict__ anchors,
    const float* __restrict__ th_buf, const float* __restrict__ maxt_buf,
    const float* __restrict__ maxi_buf,
    float* __restrict__ p2, float* __restrict__ p3, float* __restrict__ ps) {
    __shared__ float s_tgt[T_ * 5];
    __shared__ float sbuf[256];
    const int b   = blockIdx.y;
    const int a   = blockIdx.x * 256 + threadIdx.x;
    const int tid = threadIdx.x;

    for (int i = tid; i < T_ * 5; i += 256) s_tgt[i] = target[(size_t)b * T_ * 5 + i];
    __syncthreads();

    float c2 = 0.f, c3 = 0.f, cs = 0.f;
    if (a < A_) {
        const float ax = anchors[2 * a], ay = anchors[2 * a + 1];
        const float4 pb = *(const float4*)(pbox + ((size_t)b * A_ + a) * 4);
        const float* clsrow = pcls + ((size_t)b * A_ + a) * C_;

        float best = 0.f; int bi = 0;
        for (int t = 0; t < T_; ++t) {
            const float* tr = &s_tgt[t * 5];
            float tx0 = tr[1], ty0 = tr[2], tx1 = tr[3], ty1 = tr[4];
            bool gm = (tx0 < ax) & (ax < tx1) & (ty0 < ay) & (ay < ty1);
            if (!gm) continue;
            float iou_c = ciou_f(tx0, ty0, tx1, ty1, pb.x, pb.y, pb.z, pb.w);
            iou_c = fminf(fmaxf(iou_c, 0.f), 1.f);
            int c = (int)tr[0]; if (c < 0) c = 0;
            float tm = tm_value(iou_c, sigmoid_fast(clsrow[c]));
            float th = th_buf[b * T_ + t];
            float tk = (tm > 0.f && tm >= th) ? tm : 0.f;
            if (tk > best) { best = tk; bi = t; }
        }
        if (best > 0.f) {                    // valid_mask
            const int u = bi;                // uniq (first argmax)
            float s = best / (maxt_buf[b * T_ + u] + EPSF) * maxi_buf[b * T_ + u];
            const float* tr = &s_tgt[u * 5];
            int c = (int)tr[0]; if (c < 0) c = 0;
            c2 = clsrow[c] * s;              // x * t  contribution
            float iou2 = ciou_f(pb.x, pb.y, pb.z, pb.w, tr[1], tr[2], tr[3], tr[4]);
            c3 = (1.f - iou2) * s;           // box loss numerator
            cs = s;                          // box_norm / cls_norm contribution
        }
    }
    float r2 = blockReduceSum(c2, sbuf);
    float r3 = blockReduceSum(c3, sbuf);
    float rs = blockReduceSum(cs, sbuf);
    if (tid == 0) {
        int pidx = b * gridDim.x + blockIdx.x;
        p2[pidx] = r2; p3[pidx] = r3; ps[pidx] = rs;
    }
}

// ---------------------------------------------------------------- Kernel 3: softplus-BCE sum, WMMA f32
// Guard-free full-tile loop (b128 load + 4 softplus + 2 WMMA), single guarded tail
// step. Two independent V_WMMA_F32_16X16X4_F32 accumulate chains (B = ones => every
// D[m,n] carries row m's running sum; grand total = sum over lanes / 16, exact /2^4).
__global__ __launch_bounds__(256) void k_bce(const float* __restrict__ x,
                                             float* __restrict__ p1, int N) {
    const int gtid  = blockIdx.x * blockDim.x + threadIdx.x;
    const int total = gridDim.x * blockDim.x;
    const int per   = total * 4;
    const int nfull = N / per;                 // iterations where every lane is in-bounds

    v8f acc0 = {0.f, 0.f, 0.f, 0.f, 0.f, 0.f, 0.f, 0.f};
    v8f acc1 = {0.f, 0.f, 0.f, 0.f, 0.f, 0.f, 0.f, 0.f};
    v2f ones; ones[0] = 1.f; ones[1] = 1.f;

    const float* px = x + gtid * 4;
    for (int it = 0; it < nfull; ++it) {       // hot loop: no bounds checks at all
        __builtin_prefetch(px + per, 0, 0);    // global_prefetch_b8
        const float4 v = *(const float4*)px;
        v2f a0; a0[0] = softplus_bce(v.x); a0[1] = softplus_bce(v.y);
        v2f a1; a1[0] = softplus_bce(v.z); a1[1] = softplus_bce(v.w);
        acc0 = __builtin_amdgcn_wmma_f32_16x16x4_f32(false, a0, false, ones,
                                                     (short)0, acc0, false, false);
        acc1 = __builtin_amdgcn_wmma_f32_16x16x4_f32(false, a1, false, ones,
                                                     (short)0, acc1, false, false);
        px += per;
    }
    if (nfull * per < N) {                     // single tail step (uniform branch)
        const int i0 = nfull * per + gtid * 4;
        float sp0 = 0.f, sp1 = 0.f, sp2 = 0.f, sp3 = 0.f;
        if (i0     < N) sp0 = softplus_bce(x[i0]);
        if (i0 + 1 < N) sp1 = softplus_bce(x[i0 + 1]);
        if (i0 + 2 < N) sp2 = softplus_bce(x[i0 + 2]);
        if (i0 + 3 < N) sp3 = softplus_bce(x[i0 + 3]);
        v2f a0; a0[0] = sp0; a0[1] = sp1;
        v2f a1; a1[0] = sp2; a1[1] = sp3;
        // unconditional: EXEC reconverged to all-ones before the matrix ops
        acc0 = __builtin_amdgcn_wmma_f32_16x16x4_f32(false, a0, false, ones,
                                                     (short)0, acc0, false, false);
        acc1 = __builtin_amdgcn_wmma_f32_16x16x4_f32(false, a1, false, ones,
                                                     (short)0, acc1, false, false);
    }

    float l = (acc0[0] + acc0[1] + acc0[2] + acc0[3] + acc0[4] + acc0[5] + acc0[6] + acc0[7])
            + (acc1[0] + acc1[1] + acc1[2] + acc1[3] + acc1[4] + acc1[5] + acc1[6] + acc1[7]);
    #pragma unroll
    for (int o = 16; o > 0; o >>= 1) l += __shfl_xor(l, o, 32);   // wave32 reduce

    __shared__ float sw[8];
    const int lane = threadIdx.x & 31, wid = threadIdx.x >> 5;
    if (lane == 0) sw[wid] = l;
    __syncthreads();
    if (threadIdx.x == 0) {
        float t = 0.f;
        #pragma unroll
        for (int i = 0; i < 8; ++i) t += sw[i];
        p1[blockIdx.x] = t * (1.0f / 16.0f);   // 16 N-columns each replicate the total
    }
}

// ---------------------------------------------------------------- Kernel 4: final combine
__global__ __launch_bounds__(256) void k_final(const float* __restrict__ p1, int n1,
                                               const float* __restrict__ p2,
                                               const float* __restrict__ p3,
                                               const float* __restrict__ ps, int n2,
                                               float* __restrict__ out) {
    __shared__ float sbuf[256];
    const int tid = threadIdx.x;
    float a1 = 0.f, a2 = 0.f, a3 = 0.f, as = 0.f;
    for (int i = tid; i < n1; i += 256) a1 += p1[i];
    for (int i = tid; i < n2; i += 256) { a2 += p2[i]; a3 += p3[i]; as += ps[i]; }
    float S1 = blockReduceSum(a1, sbuf);
    float S2 = blockReduceSum(a2, sbuf);
    float S3 = blockReduceSum(a3, sbuf);
    float SS = blockReduceSum(as, sbuf);
    if (tid == 0) {
        float cls_norm = fmaxf(SS, 1.0f);
        out[0] = 0.5f * ((S1 - S2) / cls_norm) + 7.5f * (S3 / cls_norm);
    }
}

// ---------------------------------------------------------------- launcher
extern "C" void kernel_launch(void* const* d_in, const int* in_sizes, int n_in,
                              void* d_out, int out_size, void* d_ws, size_t ws_size,
                              hipStream_t stream) {
    const float* target  = (const float*)d_in[0];   // (16,60,5)
    const float* pcls    = (const float*)d_in[1];   // (16,8400,80)
    const float* pbox    = (const float*)d_in[2];   // (16,8400,4)
    const float* anchors = (const float*)d_in[3];   // (8400,2)

    float* ws = (float*)d_ws;                 // ~22 KB of scratch, fully rewritten per call
    float* th = ws;                           // 960
    float* mt = th + B_ * T_;                 // 960
    float* mi = mt + B_ * T_;                 // 960
    const int NBLK_A = (A_ + 255) / 256;      // 33
    const int NP2    = B_ * NBLK_A;           // 528
    float* p2 = mi + B_ * T_;                 // 528
    float* p3 = p2 + NP2;                     // 528
    float* ps = p3 + NP2;                     // 528
    const int NBCE = 1024;
    float* p1 = ps + NP2;                     // 1024

    k_rowstats<<<B_ * T_, 256, 0, stream>>>(target, pcls, pbox, anchors, th, mt, mi);

    dim3 gmatch(NBLK_A, B_);
    k_match<<<gmatch, 256, 0, stream>>>(target, pcls, pbox, anchors, th, mt, mi, p2, p3, ps);

    k_bce<<<NBCE, 256, 0, stream>>>(pcls, p1, B_ * A_ * C_);

    k_final<<<1, 256, 0, stream>>>(p1, NBCE, p2, p3, ps, NP2, (float*)d_out);
}